// CrossGraphNetLite_62577673503030
// MI455X (gfx1250) — compile-verified
//
#include <hip/hip_runtime.h>
#include <hip/hip_bf16.h>

typedef __attribute__((ext_vector_type(2))) float v2f;
typedef __attribute__((ext_vector_type(8))) float v8f;

namespace {

constexpr int N_NODES = 100000;   // N_AST == N_CFG
constexpr int N_EDGES = 1250000;  // E_AST == E_CFG
constexpr int G       = 1024;
constexpr int HID     = 64;
constexpr int SEMD    = 768;

__host__ __device__ inline int cdiv(int a, int b) { return (a + b - 1) / b; }

// ---------------- elementwise / scatter kernels ----------------

__global__ void k_gather_embed(const int* __restrict__ types, const float* __restrict__ emb,
                               float* __restrict__ x, int n) {
  int gid = blockIdx.x * blockDim.x + threadIdx.x;
  if (gid >= n * HID) return;
  int i = gid >> 6, j = gid & 63;
  x[gid] = emb[types[i] * HID + j];
}

__global__ void k_fill(float* __restrict__ p, float v, int n) {
  int gid = blockIdx.x * blockDim.x + threadIdx.x;
  if (gid < n) p[gid] = v;
}

__global__ void k_deg_accum(const int* __restrict__ dst, float* __restrict__ deg, int e) {
  int gid = blockIdx.x * blockDim.x + threadIdx.x;
  if (gid < e) atomicAdd(&deg[dst[gid]], 1.0f);
}

__global__ void k_rsqrt_inplace(float* __restrict__ p, int n) {
  int gid = blockIdx.x * blockDim.x + threadIdx.x;
  if (gid < n) p[gid] = rsqrtf(p[gid]);
}

// agg = h * dinv^2 + bias   (self-loop term + bias, full overwrite)
__global__ void k_agg_init(const float* __restrict__ h, const float* __restrict__ dinv,
                           const float* __restrict__ bias, float* __restrict__ agg, int n) {
  int gid = blockIdx.x * blockDim.x + threadIdx.x;
  if (gid >= n * HID) return;
  int i = gid >> 6, j = gid & 63;
  float di = dinv[i];
  agg[gid] = h[gid] * di * di + bias[j];
}

// agg[dst] += h[src] * dinv[src]*dinv[dst]   (one thread per edge x feature)
__global__ void k_edge_agg(const int* __restrict__ src, const int* __restrict__ dst,
                           const float* __restrict__ h, const float* __restrict__ dinv,
                           float* __restrict__ agg, int e) {
  int gid = blockIdx.x * blockDim.x + threadIdx.x;
  if (gid >= e * HID) return;
  int ei = gid >> 6, j = gid & 63;
  int s = src[ei], d = dst[ei];
  float w = dinv[s] * dinv[d];
  atomicAdd(&agg[(size_t)d * HID + j], h[(size_t)s * HID + j] * w);
}

__global__ void k_relu_inplace(float* __restrict__ p, int n) {
  int gid = blockIdx.x * blockDim.x + threadIdx.x;
  if (gid < n) p[gid] = fmaxf(p[gid], 0.0f);
}

// pooled[batch[i]] += x[i]
__global__ void k_pool(const float* __restrict__ x, const int* __restrict__ batch,
                       float* __restrict__ pooled, int n) {
  int gid = blockIdx.x * blockDim.x + threadIdx.x;
  if (gid >= n * HID) return;
  int i = gid >> 6, j = gid & 63;
  atomicAdd(&pooled[(size_t)batch[i] * HID + j], x[gid]);
}

__global__ void k_concat(const float* __restrict__ a, const float* __restrict__ b,
                         float* __restrict__ cat, int rows) {
  int gid = blockIdx.x * blockDim.x + threadIdx.x;
  if (gid >= rows * 2 * HID) return;
  int r = gid >> 7, j = gid & 127;
  cat[gid] = (j < HID) ? a[r * HID + j] : b[r * HID + (j - HID)];
}

// out = sigmoid(t)*a + (1-sigmoid(t))*b
__global__ void k_gated_mix(const float* __restrict__ t, const float* __restrict__ a,
                            const float* __restrict__ b, float* __restrict__ out, int n) {
  int gid = blockIdx.x * blockDim.x + threadIdx.x;
  if (gid >= n) return;
  float gate = 1.0f / (1.0f + __expf(-t[gid]));
  out[gid] = gate * a[gid] + (1.0f - gate) * b[gid];
}

// ---------------- WMMA fp32 GEMM: [M,K] @ [K,64] -> [M,64] ----------------
// Block = 128 threads (4 waves), each wave owns one 16-col tile; grid.x = M/16.
// Uses V_WMMA_F32_16X16X4_F32 (exact fp32, matches reference math).
template <int K, bool BIAS, bool RELU>
__global__ void k_gemm_n64(const float* __restrict__ A, const float* __restrict__ W,
                           const float* __restrict__ bias, float* __restrict__ C) {
  const int row0  = blockIdx.x << 4;
  const int lane  = threadIdx.x & 31;
  const int ncol0 = (threadIdx.x >> 5) << 4;  // wave id * 16
  const int m     = lane & 15;
  const int kh    = (lane >> 4) << 1;         // 0 or 2 (K sub-column per ISA A/B layout)

  v8f acc = {};
  const float* arow = A + (size_t)(row0 + m) * K + kh;       // A: row m, K=kb+kh, kb+kh+1
  const float* wcol = W + (size_t)kh * HID + ncol0 + m;      // B: col m, K rows kb+kh, kb+kh+1

  for (int kb = 0; kb < K; kb += 4) {
    v2f a, b;
    a.x = arow[kb];
    a.y = arow[kb + 1];
    b.x = wcol[(size_t)kb * HID];
    b.y = wcol[(size_t)(kb + 1) * HID];
    acc = __builtin_amdgcn_wmma_f32_16x16x4_f32(false, a, false, b, (short)0, acc,
                                                false, false);
  }

  const int col = ncol0 + m;
  const int rb  = row0 + ((lane >> 4) << 3);  // lanes 16-31 hold rows +8
  float bv = BIAS ? bias[col] : 0.0f;
#pragma unroll
  for (int v = 0; v < 8; ++v) {
    float val = acc[v] + bv;
    if (RELU) val = fmaxf(val, 0.0f);
    C[(size_t)(rb + v) * HID + col] = val;
  }
}

// ---------------- LayerNorm + classifier (one wave32 per row) ----------------
__global__ void k_ln_cls(const float* __restrict__ h, const float* __restrict__ g,
                         const float* __restrict__ b, const float* __restrict__ Wc,
                         const float* __restrict__ bc, float* __restrict__ out) {
  int row  = blockIdx.x * 8 + (threadIdx.x >> 5);
  int lane = threadIdx.x & 31;
  const float* hr = h + (size_t)row * HID;
  float h0 = hr[2 * lane], h1 = hr[2 * lane + 1];
  float s = h0 + h1, sq = h0 * h0 + h1 * h1;
#pragma unroll
  for (int msk = 16; msk > 0; msk >>= 1) {
    s  += __shfl_xor(s, msk, 32);
    sq += __shfl_xor(sq, msk, 32);
  }
  float mu  = s * (1.0f / 64.0f);
  float var = sq * (1.0f / 64.0f) - mu * mu;
  float rs  = rsqrtf(var + 1e-5f);
  float n0 = (h0 - mu) * rs * g[2 * lane]     + b[2 * lane];
  float n1 = (h1 - mu) * rs * g[2 * lane + 1] + b[2 * lane + 1];
  float p0 = n0 * Wc[(2 * lane) * 2 + 0] + n1 * Wc[(2 * lane + 1) * 2 + 0];
  float p1 = n0 * Wc[(2 * lane) * 2 + 1] + n1 * Wc[(2 * lane + 1) * 2 + 1];
#pragma unroll
  for (int msk = 16; msk > 0; msk >>= 1) {
    p0 += __shfl_xor(p0, msk, 32);
    p1 += __shfl_xor(p1, msk, 32);
  }
  if (lane == 0) {
    out[row * 2 + 0] = p0 + bc[0];
    out[row * 2 + 1] = p1 + bc[1];
  }
}

}  // namespace

extern "C" void kernel_launch(void* const* d_in, const int* in_sizes, int n_in,
                              void* d_out, int out_size, void* d_ws, size_t ws_size,
                              hipStream_t stream) {
  const int*   ast_type  = (const int*)d_in[0];
  const int*   ast_edge  = (const int*)d_in[1];
  const int*   ast_batch = (const int*)d_in[2];
  const int*   cfg_type  = (const int*)d_in[3];
  const int*   cfg_edge  = (const int*)d_in[4];
  const int*   cfg_batch = (const int*)d_in[5];
  const float* sem       = (const float*)d_in[6];
  const float* ast_emb   = (const float*)d_in[7];
  const float* cfg_emb   = (const float*)d_in[8];
  const float* ast_W1 = (const float*)d_in[9];  const float* ast_b1 = (const float*)d_in[10];
  const float* ast_W2 = (const float*)d_in[11]; const float* ast_b2 = (const float*)d_in[12];
  const float* cfg_W1 = (const float*)d_in[13]; const float* cfg_b1 = (const float*)d_in[14];
  const float* cfg_W2 = (const float*)d_in[15]; const float* cfg_b2 = (const float*)d_in[16];
  const float* Wg1  = (const float*)d_in[17];   const float* bg1  = (const float*)d_in[18];
  const float* Wsem = (const float*)d_in[19];   const float* bsem = (const float*)d_in[20];
  const float* Wg2  = (const float*)d_in[21];   const float* bg2  = (const float*)d_in[22];
  const float* ln_g = (const float*)d_in[23];   const float* ln_b = (const float*)d_in[24];
  const float* Wc   = (const float*)d_in[25];   const float* bc   = (const float*)d_in[26];
  float* out = (float*)d_out;

  // workspace carve-up (~80 MB; AST/CFG share the big node buffers sequentially)
  float* ws = (float*)d_ws;
  size_t o = 0;
  float* bufA    = ws + o; o += (size_t)N_NODES * HID;
  float* bufB    = ws + o; o += (size_t)N_NODES * HID;
  float* bufC    = ws + o; o += (size_t)N_NODES * HID;
  float* dinv    = ws + o; o += (size_t)N_NODES;
  float* poolA   = ws + o; o += (size_t)G * HID;
  float* poolC   = ws + o; o += (size_t)G * HID;
  float* hsem    = ws + o; o += (size_t)G * HID;
  float* hstruct = ws + o; o += (size_t)G * HID;
  float* hfin    = ws + o; o += (size_t)G * HID;
  float* catbuf  = ws + o; o += (size_t)G * 2 * HID;
  float* gatet   = ws + o; o += (size_t)G * HID;

  const int TB = 256;
  const int nfeat = N_NODES * HID;   // 6.4M
  const int efeat = N_EDGES * HID;   // 80M

  auto run_gnn = [&](const int* types, const float* emb, const int* edge, const int* batch,
                     const float* W1, const float* b1, const float* W2, const float* b2,
                     float* pooled) {
    const int* src = edge;
    const int* dst = edge + N_EDGES;
    // x = emb[types]
    k_gather_embed<<<cdiv(nfeat, TB), TB, 0, stream>>>(types, emb, bufA, N_NODES);
    // dinv = (deg+1)^{-1/2}
    k_fill<<<cdiv(N_NODES, TB), TB, 0, stream>>>(dinv, 1.0f, N_NODES);
    k_deg_accum<<<cdiv(N_EDGES, TB), TB, 0, stream>>>(dst, dinv, N_EDGES);
    k_rsqrt_inplace<<<cdiv(N_NODES, TB), TB, 0, stream>>>(dinv, N_NODES);
    // layer 1: h = x@W1 ; agg = scatter(h) + h*dinv^2 + b1 ; relu
    k_gemm_n64<HID, false, false><<<N_NODES / 16, 128, 0, stream>>>(bufA, W1, nullptr, bufB);
    k_agg_init<<<cdiv(nfeat, TB), TB, 0, stream>>>(bufB, dinv, b1, bufC, N_NODES);
    k_edge_agg<<<cdiv(efeat, TB), TB, 0, stream>>>(src, dst, bufB, dinv, bufC, N_EDGES);
    k_relu_inplace<<<cdiv(nfeat, TB), TB, 0, stream>>>(bufC, nfeat);
    // layer 2
    k_gemm_n64<HID, false, false><<<N_NODES / 16, 128, 0, stream>>>(bufC, W2, nullptr, bufB);
    k_agg_init<<<cdiv(nfeat, TB), TB, 0, stream>>>(bufB, dinv, b2, bufA, N_NODES);
    k_edge_agg<<<cdiv(efeat, TB), TB, 0, stream>>>(src, dst, bufB, dinv, bufA, N_EDGES);
    // pool
    k_fill<<<cdiv(G * HID, TB), TB, 0, stream>>>(pooled, 0.0f, G * HID);
    k_pool<<<cdiv(nfeat, TB), TB, 0, stream>>>(bufA, batch, pooled, N_NODES);
  };

  run_gnn(ast_type, ast_emb, ast_edge, ast_batch, ast_W1, ast_b1, ast_W2, ast_b2, poolA);
  run_gnn(cfg_type, cfg_emb, cfg_edge, cfg_batch, cfg_W1, cfg_b1, cfg_W2, cfg_b2, poolC);

  // h_sem = relu(sem @ Wsem + bsem)
  k_gemm_n64<SEMD, true, true><<<G / 16, 128, 0, stream>>>(sem, Wsem, bsem, hsem);

  // gated fuse 1: struct = g*ast + (1-g)*cfg
  k_concat<<<cdiv(G * 2 * HID, TB), TB, 0, stream>>>(poolA, poolC, catbuf, G);
  k_gemm_n64<2 * HID, true, false><<<G / 16, 128, 0, stream>>>(catbuf, Wg1, bg1, gatet);
  k_gated_mix<<<cdiv(G * HID, TB), TB, 0, stream>>>(gatet, poolA, poolC, hstruct, G * HID);

  // gated fuse 2: h = g*struct + (1-g)*sem
  k_concat<<<cdiv(G * 2 * HID, TB), TB, 0, stream>>>(hstruct, hsem, catbuf, G);
  k_gemm_n64<2 * HID, true, false><<<G / 16, 128, 0, stream>>>(catbuf, Wg2, bg2, gatet);
  k_gated_mix<<<cdiv(G * HID, TB), TB, 0, stream>>>(gatet, hstruct, hsem, hfin, G * HID);

  // LayerNorm + classifier
  k_ln_cls<<<G / 8, 256, 0, stream>>>(hfin, ln_g, ln_b, Wc, bc, out);
}